// CustomReshapeLayer_55619826483452
// MI455X (gfx1250) — compile-verified
//
#include <hip/hip_runtime.h>

// Problem constants (from the reference)
#define MATSIZE 2048
#define NTRIU   (MATSIZE * (MATSIZE + 1) / 2)   // 2,098,176
#define NBATCH  32

typedef float v4f __attribute__((ext_vector_type(4)));

// Gather formulation of the triu scatter:
//   out[b, m, c] = (c >= m) ? in[b*NTRIU + base(m) + (c - m)] : 0
// with base(m) = m*MATSIZE - m*(m-1)/2.
//
// Pure streaming kernel: ~805 MB of touch-once traffic -> HBM-bound
// (~35 us floor at 23.3 TB/s). All stores are 16B-aligned b128 with
// non-temporal hint; loads are non-temporal b32 (coalesced per row).
__global__ __launch_bounds__(256) void triu_to_dense_kernel(
    const float* __restrict__ in, float* __restrict__ out) {
    const int m  = blockIdx.y;                               // matrix row
    const int b  = blockIdx.z;                               // batch
    const int c0 = (blockIdx.x * 256 + threadIdx.x) * 4;     // first of 4 columns

    // Flat-triu row start for row m (column m maps here).
    const size_t rowBase = (size_t)m * MATSIZE - ((size_t)m * (m - 1)) / 2;
    // Shifted pointer so inRow[c] is the value for column c (valid for c >= m).
    const float* __restrict__ inRow =
        in + (size_t)b * NTRIU + rowBase - (size_t)m;
    float* __restrict__ outPtr =
        out + (((size_t)b * MATSIZE + m) * MATSIZE + c0);

    v4f v;
    if (c0 >= m) {
        // Fully inside the upper triangle: 4 streaming (NT) loads, coalesced.
        v.x = __builtin_nontemporal_load(inRow + c0 + 0);
        v.y = __builtin_nontemporal_load(inRow + c0 + 1);
        v.z = __builtin_nontemporal_load(inRow + c0 + 2);
        v.w = __builtin_nontemporal_load(inRow + c0 + 3);
        // Stay ahead of the read stream (lowers to global_prefetch_b8).
        __builtin_prefetch(inRow + c0 + 1024, 0, 0);
    } else if (c0 + 3 < m) {
        // Fully strictly-lower-triangular: zeros.
        v = (v4f){0.0f, 0.0f, 0.0f, 0.0f};
    } else {
        // The one mixed 4-wide group per row straddling the diagonal.
        #pragma unroll
        for (int j = 0; j < 4; ++j) {
            const int c = c0 + j;
            v[j] = (c >= m) ? inRow[c] : 0.0f;
        }
    }
    // Aligned 16B non-temporal store: global_store_b128 with TH=NT.
    __builtin_nontemporal_store(v, (v4f*)outPtr);
}

extern "C" void kernel_launch(void* const* d_in, const int* in_sizes, int n_in,
                              void* d_out, int out_size, void* d_ws, size_t ws_size,
                              hipStream_t stream) {
    (void)in_sizes; (void)n_in; (void)out_size; (void)d_ws; (void)ws_size;
    const float* in = (const float*)d_in[0];   // [32, NTRIU] f32
    float* out      = (float*)d_out;           // [32, 2048, 2048] f32

    // 256 threads (8 wave32) x 4 floats = 1024 columns per block -> 2 blocks/row.
    dim3 block(256, 1, 1);
    dim3 grid(MATSIZE / (256 * 4), MATSIZE, NBATCH);  // (2, 2048, 32)
    triu_to_dense_kernel<<<grid, block, 0, stream>>>(in, out);
}